// GCN_8495445311789
// MI455X (gfx1250) — compile-verified
//
#include <hip/hip_runtime.h>

#define N_NODES 100000
#define N_EDGES 500000
#define DIM     128
#define SLOPE   0.01f
#define LDW     132      // LDS row stride (floats): 132*4 B -> lanes land on distinct banks
#define MT      10       // M-tiles (16 rows) per block in the GEMM

typedef float v2f __attribute__((ext_vector_type(2)));
typedef float v8f __attribute__((ext_vector_type(8)));

// ---------------------------------------------------------------------------
// Async global->LDS helpers (CDNA5 ASYNCcnt path)
// ---------------------------------------------------------------------------
__device__ __forceinline__ void async_copy_b128(unsigned lds_off, const void* gptr) {
    asm volatile("global_load_async_to_lds_b128 %0, %1, off"
                 :: "v"(lds_off), "v"((unsigned long long)(__SIZE_TYPE__)gptr)
                 : "memory");
}

template <int N>
__device__ __forceinline__ void wait_async() {
#if __has_builtin(__builtin_amdgcn_s_wait_asynccnt)
    __builtin_amdgcn_s_wait_asynccnt(N);
#else
    asm volatile("s_wait_asynccnt %0" :: "i"(N) : "memory");
#endif
}

__device__ __forceinline__ unsigned lds_off_of(const void* p) {
    // generic pointer to __shared__: low 32 bits are the LDS byte offset
    return (unsigned)(unsigned long long)p;
}

// ---------------------------------------------------------------------------
// ws layout (floats):
//   [0,128) v_l ; [128,256) v_r ; al[N] ; ar[N] ; abuf[E] ; 16 scalars ; acc[N*D]
// ---------------------------------------------------------------------------

// Kernel 1: v_l = W_at^T @ w_att_l, v_r = W_at^T @ w_att_r   (z-GEMM collapsed)
__global__ void gat_prep_vec(const float* __restrict__ W_at,
                             const float* __restrict__ wl,
                             const float* __restrict__ wr,
                             float* __restrict__ vlr) {
    int k = threadIdx.x;
    float sl = 0.f, sr = 0.f;
    for (int j = 0; j < DIM; ++j) {
        float w = W_at[j * DIM + k];
        sl += w * wl[j];
        sr += w * wr[j];
    }
    vlr[k]       = sl;
    vlr[DIM + k] = sr;
}

// Kernel 2: one wave32 per node row: al/ar dot products
__global__ void gat_node_logits(const float* __restrict__ feat,
                                const float* __restrict__ vlr,
                                float* __restrict__ al,
                                float* __restrict__ ar) {
    int wave = threadIdx.x >> 5;
    int lane = threadIdx.x & 31;
    int row  = blockIdx.x * 8 + wave;
    if (row >= N_NODES) return;
    const float* f = feat + row * DIM;
    float sl = 0.f, sr = 0.f;
#pragma unroll
    for (int i = 0; i < 4; ++i) {
        int c   = lane + i * 32;
        float v = f[c];
        sl += v * vlr[c];
        sr += v * vlr[DIM + c];
    }
#pragma unroll
    for (int off = 16; off > 0; off >>= 1) {
        sl += __shfl_down(sl, off, 32);
        sr += __shfl_down(sr, off, 32);
    }
    if (lane == 0) { al[row] = sl; ar[row] = sr; }
}

// Kernel 3: zero accumulator + reduction scalars
__global__ void gat_init(float* __restrict__ acc,
                         unsigned int* __restrict__ gmax,
                         float* __restrict__ gsum) {
    int t = blockIdx.x * 256 + threadIdx.x;
    if (t < N_NODES * DIM) acc[t] = 0.f;
    if (t == 0) { *gmax = 0u; *gsum = 0.f; }
}

// Kernel 4: edge logits + leaky-relu + global max (monotonic uint keys)
__global__ void gat_edge_logits(const int* __restrict__ src,
                                const int* __restrict__ dst,
                                const float* __restrict__ al,
                                const float* __restrict__ ar,
                                float* __restrict__ abuf,
                                unsigned int* __restrict__ gmax) {
    __shared__ unsigned int smax;
    int t = blockIdx.x * 256 + threadIdx.x;
    if (threadIdx.x == 0) smax = 0u;
    __syncthreads();
    if (t < N_EDGES) {
        float a = al[src[t]] + ar[dst[t]];
        a = (a > 0.f) ? a : SLOPE * a;
        abuf[t] = a;
        unsigned int bits = __float_as_uint(a);
        unsigned int key  = (bits & 0x80000000u) ? ~bits : (bits | 0x80000000u);
        atomicMax(&smax, key);
    }
    __syncthreads();
    if (threadIdx.x == 0) atomicMax(gmax, smax);
}

// Kernel 5: exp(a - max) in place + global sum
__global__ void gat_edge_exp(float* __restrict__ abuf,
                             const unsigned int* __restrict__ gmax,
                             float* __restrict__ gsum) {
    __shared__ float ssum;
    int t = blockIdx.x * 256 + threadIdx.x;
    if (threadIdx.x == 0) ssum = 0.f;
    __syncthreads();
    unsigned int key  = *gmax;
    unsigned int bits = (key & 0x80000000u) ? (key & 0x7fffffffu) : ~key;
    float mx = __uint_as_float(bits);
    if (t < N_EDGES) {
        float v = __expf(abuf[t] - mx);
        abuf[t] = v;
        atomicAdd(&ssum, v);
    }
    __syncthreads();
    if (threadIdx.x == 0) atomicAdd(gsum, ssum);
}

// Kernel 6: scatter-add aggregation. One wave per edge (uniform e -> scalar
// loads for src/dst/weight), float4 per lane -> coalesced b128 gathers.
__global__ void gat_aggregate(const int* __restrict__ src,
                              const int* __restrict__ dst,
                              const float* __restrict__ feat,
                              const float* __restrict__ abuf,
                              float* __restrict__ acc) {
    int e = blockIdx.x * 8 + (threadIdx.x >> 5);   // uniform per wave
    int q = threadIdx.x & 31;
    float w = abuf[e];
    const float4 f = ((const float4*)feat)[src[e] * (DIM / 4) + q];
    float* p = acc + dst[e] * DIM + q * 4;
    atomicAdd(p + 0, f.x * w);
    atomicAdd(p + 1, f.y * w);
    atomicAdd(p + 2, f.z * w);
    atomicAdd(p + 3, f.w * w);
}

// Kernel 7: out = (feat + acc/sum) @ W_self^T + b_self with fp32 WMMA.
// 625 blocks x MT=10 row-tiles: W_self staged to LDS ONCE per block (async),
// B fragments hoisted to registers, feat/acc tiles double-buffered via
// GLOBAL_LOAD_ASYNC_TO_LDS_B128 + s_wait_asynccnt pipelining.
__global__ __launch_bounds__(256) void gat_self_gemm(
        const float* __restrict__ feat,
        const float* __restrict__ W_self,
        const float* __restrict__ b_self,
        const float* __restrict__ acc,
        const float* __restrict__ gsum,
        float* __restrict__ out) {
    __shared__ float Wl[128 * LDW];     // full W_self, padded rows
    __shared__ float Fb[2][16 * LDW];   // feat tile ping-pong
    __shared__ float Ab[2][16 * LDW];   // acc tile ping-pong

    const int tid       = threadIdx.x;
    const int base_tile = blockIdx.x * MT;
    const float inv     = 1.0f / gsum[0];

    // ---- async-stage all of W_self: 16 b128 per thread ----
#pragma unroll
    for (int i = 0; i < 16; ++i) {
        int q = i * 256 + tid;                 // float4 index 0..4095
        int r = q >> 5, c = q & 31;
        async_copy_b128(lds_off_of(&Wl[r * LDW + 4 * c]), W_self + q * 4);
    }
    // ---- async-stage tile 0 (feat + acc): 4 b128 per thread ----
    {
#pragma unroll
        for (int i = 0; i < 2; ++i) {
            int q = i * 256 + tid;             // float4 index 0..511
            int r = q >> 5, c = q & 31;
            int grow = base_tile * 16 + r;
            async_copy_b128(lds_off_of(&Fb[0][r * LDW + 4 * c]),
                            feat + grow * DIM + 4 * c);
            async_copy_b128(lds_off_of(&Ab[0][r * LDW + 4 * c]),
                            acc + grow * DIM + 4 * c);
        }
    }
    wait_async<4>();        // in-order: all 16 W loads done; tile0 still in flight
    __syncthreads();        // W_self visible to every wave

    const int wave = tid >> 5;
    const int lane = tid & 31;
    const int nl   = lane & 15;
    const int koff = (lane < 16) ? 0 : 2;      // ISA fp32 16x4 lane split
    const int n0   = wave * 16;
    const int col  = n0 + nl;
    const float bs = b_self[col];

    // hoist B fragments (this wave's 16 columns of W_self^T) into registers
    const float* Wrow = &Wl[(n0 + nl) * LDW];
    v2f B[32];
#pragma unroll
    for (int k4 = 0; k4 < 32; ++k4)
        B[k4] = *(const v2f*)&Wrow[4 * k4 + koff];

    for (int t = 0; t < MT; ++t) {
        const int buf = t & 1;
        // kick tile t+1 into the other buffer
        if (t + 1 < MT) {
#pragma unroll
            for (int i = 0; i < 2; ++i) {
                int q = i * 256 + tid;
                int r = q >> 5, c = q & 31;
                int grow = (base_tile + t + 1) * 16 + r;
                async_copy_b128(lds_off_of(&Fb[buf ^ 1][r * LDW + 4 * c]),
                                feat + grow * DIM + 4 * c);
                async_copy_b128(lds_off_of(&Ab[buf ^ 1][r * LDW + 4 * c]),
                                acc + grow * DIM + 4 * c);
            }
            wait_async<4>();   // retire tile t's 4 requests (in-order)
        } else {
            wait_async<0>();
        }
        __syncthreads();        // tile t fully resident for all waves

        const float* Fl = &Fb[buf][nl * LDW];
        const float* Al = &Ab[buf][nl * LDW];
        v8f cacc = {0.f, 0.f, 0.f, 0.f, 0.f, 0.f, 0.f, 0.f};
#pragma unroll
        for (int k4 = 0; k4 < 32; ++k4) {
            v2f f = *(const v2f*)&Fl[4 * k4 + koff];
            v2f a = *(const v2f*)&Al[4 * k4 + koff];
            v2f tt;
            tt.x = fmaf(a.x, inv, f.x);        // t = feat + acc/sum
            tt.y = fmaf(a.y, inv, f.y);
            cacc = __builtin_amdgcn_wmma_f32_16x16x4_f32(
                false, tt, false, B[k4], (short)0, cacc, false, false);
        }

        const int rb = (base_tile + t) * 16 + ((lane < 16) ? 0 : 8);
#pragma unroll
        for (int r = 0; r < 8; ++r)
            out[(rb + r) * DIM + col] = cacc[r] + bs;

        __syncthreads();        // protect buf before t+2 overwrites it
    }
}

extern "C" void kernel_launch(void* const* d_in, const int* in_sizes, int n_in,
                              void* d_out, int out_size, void* d_ws, size_t ws_size,
                              hipStream_t stream) {
    (void)in_sizes; (void)n_in; (void)out_size; (void)ws_size;
    const float* feat   = (const float*)d_in[0];
    const float* W_at   = (const float*)d_in[1];
    const float* wl     = (const float*)d_in[2];
    const float* wr     = (const float*)d_in[3];
    const float* W_self = (const float*)d_in[4];
    const float* b_self = (const float*)d_in[5];
    const int*   src    = (const int*)d_in[6];
    const int*   dst    = (const int*)d_in[7];
    float* out = (float*)d_out;
    float* ws  = (float*)d_ws;

    float* vlr  = ws;                         // 256
    float* al   = ws + 256;                   // N
    float* ar   = al + N_NODES;               // N
    float* abuf = ar + N_NODES;               // E
    float* scal = abuf + N_EDGES;             // 16 scalar slots
    unsigned int* gmax = (unsigned int*)scal;
    float* gsum = scal + 1;
    float* acc  = scal + 16;                  // N*D (16-float aligned offset)

    gat_prep_vec   <<<1, 128, 0, stream>>>(W_at, wl, wr, vlr);
    gat_node_logits<<<N_NODES / 8, 256, 0, stream>>>(feat, vlr, al, ar);
    gat_init       <<<(N_NODES * DIM + 255) / 256, 256, 0, stream>>>(acc, gmax, gsum);
    gat_edge_logits<<<(N_EDGES + 255) / 256, 256, 0, stream>>>(src, dst, al, ar, abuf, gmax);
    gat_edge_exp   <<<(N_EDGES + 255) / 256, 256, 0, stream>>>(abuf, gmax, gsum);
    gat_aggregate  <<<N_EDGES / 8, 256, 0, stream>>>(src, dst, feat, abuf, acc);
    gat_self_gemm  <<<N_NODES / (16 * MT), 256, 0, stream>>>(feat, W_self, b_self, acc, gsum, out);
}